// PointPillars_21534966022579
// MI455X (gfx1250) — compile-verified
//
#include <hip/hip_runtime.h>
#include <math.h>

typedef float v2f __attribute__((ext_vector_type(2)));
typedef float v8f __attribute__((ext_vector_type(8)));

// ---------------- problem constants ----------------
constexpr int Hh = 496, Ww = 432, NA = 6, NCls = 3;
constexpr int HW = Hh * Ww;          // 214272
constexpr int N  = HW * NA;          // 1285632 anchors
constexpr int NPRE = 4096;           // NMS_PRE
constexpr int MAXOUT = 500;
constexpr int NBUCK = 2048;
constexpr int CANDCAP = 8192;
constexpr int MASKW = NPRE / 32;     // 128 words per row
constexpr int CHUNK = 128;           // greedy-scan chunk size
#define SCORE_THR 0.1f
#define NMS_THR   0.5f
#define PI_F      3.14159265358979323846f

// ---------------- workspace layout (bytes) ----------------
constexpr size_t O_MAXS = 0;                                   // N f32
constexpr size_t O_HIST = O_MAXS + (size_t)N * 4;              // 2048 i32
constexpr size_t O_META = O_HIST + (size_t)NBUCK * 4;          // 4 i32
constexpr size_t O_CNT  = O_META + 16;                         // 4 i32
constexpr size_t O_CAND = O_CNT + 16;                          // 8192 u64
constexpr size_t O_TOPK = O_CAND + (size_t)CANDCAP * 8;        // 4096 i32
constexpr size_t O_SC3  = O_TOPK + (size_t)NPRE * 4;           // 4096*3 f32
constexpr size_t O_DIR  = O_SC3 + (size_t)NPRE * 3 * 4;        // 4096 i32
constexpr size_t O_BOX  = O_DIR + (size_t)NPRE * 4;            // 4096*7 f32
constexpr size_t O_X1   = O_BOX + (size_t)NPRE * 7 * 4;
constexpr size_t O_Y1   = O_X1 + (size_t)NPRE * 4;
constexpr size_t O_X2   = O_Y1 + (size_t)NPRE * 4;
constexpr size_t O_Y2   = O_X2 + (size_t)NPRE * 4;
constexpr size_t O_AR   = O_Y2 + (size_t)NPRE * 4;
constexpr size_t O_ORD  = O_AR + (size_t)NPRE * 4;             // sorted->orig idx
constexpr size_t O_SSC  = O_ORD + (size_t)NPRE * 4;
constexpr size_t O_SX1  = O_SSC + (size_t)NPRE * 4;
constexpr size_t O_SY1  = O_SX1 + (size_t)NPRE * 4;
constexpr size_t O_SX2  = O_SY1 + (size_t)NPRE * 4;
constexpr size_t O_SY2  = O_SX2 + (size_t)NPRE * 4;
constexpr size_t O_SAR  = O_SY2 + (size_t)NPRE * 4;
constexpr size_t O_KEEP = O_SAR + (size_t)NPRE * 4;            // 4096 i32
constexpr size_t O_MASK = O_KEEP + (size_t)NPRE * 4;           // 4096*128 u32 = 2MB
constexpr size_t O_COMB = O_MASK + (size_t)NPRE * MASKW * 4;   // 12288 f32
constexpr size_t O_T500 = O_COMB + (size_t)NPRE * NCls * 4;    // 512 i32

__device__ __forceinline__ float sigmoidf_(float x) { return 1.f / (1.f + expf(-x)); }

// float -> order-preserving u32 (handles negatives / -inf)
__device__ __forceinline__ unsigned f2s(float f) {
  unsigned u = __float_as_uint(f);
  return (u & 0x80000000u) ? ~u : (u | 0x80000000u);
}

// descending bitonic sort of n (power of two) u64 keys in LDS
__device__ __forceinline__ void bitonic_desc(unsigned long long* s, int n, int tid, int nt) {
  for (int k = 2; k <= n; k <<= 1) {
    for (int j = k >> 1; j > 0; j >>= 1) {
      for (int i = tid; i < n; i += nt) {
        int ixj = i ^ j;
        if (ixj > i) {
          unsigned long long a = s[i], b = s[ixj];
          bool desc = ((i & k) == 0);
          bool sw = desc ? (a < b) : (a > b);
          if (sw) { s[i] = b; s[ixj] = a; }
        }
      }
      __syncthreads();
    }
  }
}

// ---------------- kernels ----------------

__global__ void pp_init(unsigned* hist, unsigned* cnt, unsigned long long* cand) {
  int i = blockIdx.x * blockDim.x + threadIdx.x;
  if (i < NBUCK) hist[i] = 0u;
  if (i == 0) cnt[0] = 0u;
  if (i < CANDCAP) cand[i] = 0ull;
}

// pass 1: per-anchor max sigmoid score + histogram
__global__ void pp_score_hist(const float* __restrict__ cls, float* __restrict__ maxs,
                              unsigned* __restrict__ hist) {
  int i = blockIdx.x * blockDim.x + threadIdx.x;
  if (i >= N) return;
  int loc = i / NA, a = i % NA;
  float m = -1.f;
  #pragma unroll
  for (int c = 0; c < NCls; ++c) {
    float s = sigmoidf_(cls[(size_t)(a * NCls + c) * HW + loc]);
    m = fmaxf(m, s);
  }
  maxs[i] = m;
  int b = (int)(m * (float)NBUCK);
  b = min(max(b, 0), NBUCK - 1);
  atomicAdd(&hist[b], 1u);
}

__global__ void pp_threshold(const unsigned* __restrict__ hist, int* __restrict__ meta) {
  if (threadIdx.x == 0 && blockIdx.x == 0) {
    unsigned cum = 0; int bsel = 0;
    for (int b = NBUCK - 1; b >= 0; --b) {
      cum += hist[b];
      if (cum >= (unsigned)NPRE) { bsel = b; break; }
    }
    meta[0] = bsel;
  }
}

__global__ void pp_compact(const float* __restrict__ maxs, const int* __restrict__ meta,
                           unsigned* __restrict__ cnt, unsigned long long* __restrict__ cand) {
  int i = blockIdx.x * blockDim.x + threadIdx.x;
  if (i >= N) return;
  float m = maxs[i];
  int b = (int)(m * (float)NBUCK);
  b = min(max(b, 0), NBUCK - 1);
  if (b >= meta[0]) {
    unsigned p = atomicAdd(cnt, 1u);
    if (p < (unsigned)CANDCAP)
      cand[p] = ((unsigned long long)__float_as_uint(m) << 32) | (unsigned)(~(unsigned)i);
  }
}

__global__ __launch_bounds__(1024)
void pp_sort_cand(const unsigned long long* __restrict__ cand, int* __restrict__ topk) {
  extern __shared__ unsigned long long sm[];
  int tid = threadIdx.x;
  for (int i = tid; i < CANDCAP; i += 1024) sm[i] = cand[i];
  __syncthreads();
  bitonic_desc(sm, CANDCAP, tid, 1024);
  for (int i = tid; i < NPRE; i += 1024)
    topk[i] = (int)(~(unsigned)(sm[i] & 0xFFFFFFFFull));
}

// gather + sigmoid + dir argmax + box decode + xyxy/area for the 4096 selected
__global__ void pp_decode(const float* __restrict__ cls, const float* __restrict__ bp,
                          const float* __restrict__ dir, const float* __restrict__ pri,
                          const int* __restrict__ topk,
                          float* __restrict__ sc3, int* __restrict__ dcls,
                          float* __restrict__ box,
                          float* __restrict__ X1, float* __restrict__ Y1,
                          float* __restrict__ X2, float* __restrict__ Y2,
                          float* __restrict__ AR) {
  int i = blockIdx.x * blockDim.x + threadIdx.x;
  if (i >= NPRE) return;
  int idx = topk[i];
  if ((unsigned)idx >= (unsigned)N) idx = 0;
  int loc = idx / NA, a = idx % NA;
  #pragma unroll
  for (int c = 0; c < NCls; ++c)
    sc3[i * NCls + c] = sigmoidf_(cls[(size_t)(a * NCls + c) * HW + loc]);
  float d0 = dir[(size_t)(a * 2 + 0) * HW + loc];
  float d1 = dir[(size_t)(a * 2 + 1) * HW + loc];
  dcls[i] = (d1 > d0) ? 1 : 0;
  float t[7];
  #pragma unroll
  for (int k = 0; k < 7; ++k) t[k] = bp[(size_t)(a * 7 + k) * HW + loc];
  float xa = pri[(size_t)idx * 7 + 0], ya = pri[(size_t)idx * 7 + 1], za = pri[(size_t)idx * 7 + 2];
  float wa = pri[(size_t)idx * 7 + 3], la = pri[(size_t)idx * 7 + 4], ha = pri[(size_t)idx * 7 + 5];
  float ra = pri[(size_t)idx * 7 + 6];
  za += ha * 0.5f;
  float diag = sqrtf(la * la + wa * wa);
  float xg = t[0] * diag + xa;
  float yg = t[1] * diag + ya;
  float zg = t[2] * ha + za;
  float wg = expf(t[3]) * wa;
  float lg = expf(t[4]) * la;
  float hg = expf(t[5]) * ha;
  float rg = t[6] + ra;
  zg -= hg * 0.5f;
  box[i * 7 + 0] = xg; box[i * 7 + 1] = yg; box[i * 7 + 2] = zg;
  box[i * 7 + 3] = wg; box[i * 7 + 4] = lg; box[i * 7 + 5] = hg; box[i * 7 + 6] = rg;
  float x1 = xg - wg * 0.5f, y1 = yg - lg * 0.5f;
  float x2 = xg + wg * 0.5f, y2 = yg + lg * 0.5f;
  X1[i] = x1; Y1[i] = y1; X2[i] = x2; Y2[i] = y2;
  AR[i] = (x2 - x1 + 1.f) * (y2 - y1 + 1.f);
}

// per-class: sort 4096 boxes by (score desc, idx asc); gather sorted box data
__global__ __launch_bounds__(1024)
void pp_sort_class(const float* __restrict__ sc3,
                   const float* __restrict__ X1, const float* __restrict__ Y1,
                   const float* __restrict__ X2, const float* __restrict__ Y2,
                   const float* __restrict__ AR,
                   int* __restrict__ ord, float* __restrict__ ssc,
                   float* __restrict__ sx1, float* __restrict__ sy1,
                   float* __restrict__ sx2, float* __restrict__ sy2,
                   float* __restrict__ sar, int cls) {
  __shared__ unsigned long long sm[NPRE];   // 32 KB
  int tid = threadIdx.x;
  for (int i = tid; i < NPRE; i += 1024) {
    float s = sc3[i * NCls + cls];          // sigmoid > 0, bits are order-preserving
    sm[i] = ((unsigned long long)__float_as_uint(s) << 32) | (unsigned)(~(unsigned)i);
  }
  __syncthreads();
  bitonic_desc(sm, NPRE, tid, 1024);
  for (int p = tid; p < NPRE; p += 1024) {
    int j = (int)((~(unsigned)(sm[p] & 0xFFFFFFFFull)) & (NPRE - 1));
    ord[p] = j;
    ssc[p] = sc3[j * NCls + cls];
    sx1[p] = X1[j]; sy1[p] = Y1[j]; sx2[p] = X2[j]; sy2[p] = Y2[j]; sar[p] = AR[j];
  }
}

// One wave per 16x32 IoU tile: rows tr*16..+15, cols tc*32..+31.
// Union denominator (area_i + area_j) per 16x16 half-tile is a rank-2
// outer-sum via V_WMMA_F32_16X16X4_F32 (two issues per wave):
//   A(16x4) cols = [area_row, 1, 0, 0];  B(4x16) rows = [1; area_col; 0; 0]
// Each wave produces complete 32-bit mask words -> plain stores, no atomics,
// no zero-fill pass.
__global__ __launch_bounds__(32)
void pp_nms_mask(const float* __restrict__ sx1, const float* __restrict__ sy1,
                 const float* __restrict__ sx2, const float* __restrict__ sy2,
                 const float* __restrict__ sar, unsigned* __restrict__ mask) {
  int tc = blockIdx.x;   // 0..127 (32-column groups)
  int tr = blockIdx.y;   // 0..255 (16-row groups)
  int lane = threadIdx.x;
  __shared__ float R[5][16], C[5][32];
  {
    int c = tc * 32 + lane;
    C[0][lane] = sx1[c]; C[1][lane] = sy1[c]; C[2][lane] = sx2[c];
    C[3][lane] = sy2[c]; C[4][lane] = sar[c];
    if (lane < 16) {
      int r = tr * 16 + lane;
      R[0][lane] = sx1[r]; R[1][lane] = sy1[r]; R[2][lane] = sx2[r];
      R[3][lane] = sy2[r]; R[4][lane] = sar[r];
    }
  }
  __syncthreads();

  bool lo16 = (lane < 16);
  int n = lane & 15;
  // A: 16x4, K columns [area_row, 1, 0, 0]
  v2f Amat;
  Amat.x = lo16 ? R[4][n] : 0.f;      // K=0
  Amat.y = lo16 ? 1.f : 0.f;          // K=1 (lanes>=16 hold K=2,3 -> zero)
  // B: 4x16, K rows [1; area_col; 0; 0] for each 16-column half
  v2f Ba, Bb;
  Ba.x = lo16 ? 1.f : 0.f;
  Ba.y = lo16 ? C[4][n] : 0.f;
  Bb.x = Ba.x;
  Bb.y = lo16 ? C[4][n + 16] : 0.f;
  v8f Z = {};
  v8f Da = __builtin_amdgcn_wmma_f32_16x16x4_f32(false, Amat, false, Ba,
                                                 (short)0, Z, false, false);
  v8f Db = __builtin_amdgcn_wmma_f32_16x16x4_f32(false, Amat, false, Bb,
                                                 (short)0, Z, false, false);

  int hi = lane >> 4;  // C/D layout: lanes 0-15 -> rows v, lanes 16-31 -> rows v+8
  float ax1 = C[0][n],      ay1 = C[1][n],      ax2 = C[2][n],      ay2 = C[3][n];
  float bx1 = C[0][n + 16], by1 = C[1][n + 16], bx2 = C[2][n + 16], by2 = C[3][n + 16];
  #pragma unroll
  for (int v = 0; v < 8; ++v) {
    int m = v + 8 * hi;
    float rx1 = R[0][m], ry1 = R[1][m], rx2 = R[2][m], ry2 = R[3][m];
    float iax = fminf(rx2, ax2) - fmaxf(rx1, ax1) + 1.f;
    float iay = fminf(ry2, ay2) - fmaxf(ry1, ay1) + 1.f;
    float interA = fmaxf(iax, 0.f) * fmaxf(iay, 0.f);
    float ibx = fminf(rx2, bx2) - fmaxf(rx1, bx1) + 1.f;
    float iby = fminf(ry2, by2) - fmaxf(ry1, by1) + 1.f;
    float interB = fmaxf(ibx, 0.f) * fmaxf(iby, 0.f);
    bool supA = interA > NMS_THR * (Da[v] - interA);   // iou > thr
    bool supB = interB > NMS_THR * (Db[v] - interB);
    unsigned balA = (unsigned)__ballot(supA);          // wave32: 32 meaningful bits
    unsigned balB = (unsigned)__ballot(supB);
    unsigned wordLo = (balA & 0xFFFFu) | ((balB & 0xFFFFu) << 16);  // row v
    unsigned wordHi = (balA >> 16)     | (balB & 0xFFFF0000u);      // row v+8
    if (lane == v)      mask[(size_t)(tr * 16 + v)     * MASKW + tc] = wordLo;
    if (lane == v + 16) mask[(size_t)(tr * 16 + v + 8) * MASKW + tc] = wordHi;
  }
}

// chunked greedy scan: serial dependency is per-128-box chunk (32 steps),
// intra-chunk decisions run on LDS only; kept rows are OR-ed in a batched,
// memory-parallel pass.
__global__ __launch_bounds__(CHUNK)
void pp_nms_scan(const float* __restrict__ ssc, const int* __restrict__ ord,
                 const unsigned* __restrict__ mask, int* __restrict__ keep,
                 float* __restrict__ comb, int cls) {
  __shared__ unsigned sup[MASKW];           // global suppression bitset (4096 bits)
  __shared__ unsigned intra[CHUNK * 4];     // per-row words covering this chunk
  __shared__ unsigned keptw[4];             // kept bitset for current chunk
  __shared__ float cscore[CHUNK];
  int tid = threadIdx.x;
  sup[tid] = 0u;
  __syncthreads();
  for (int chunk = 0; chunk < NPRE / CHUNK; ++chunk) {
    int base = chunk * CHUNK;
    int wbase = base >> 5;                  // 4 words span this chunk's columns
    #pragma unroll
    for (int q = 0; q < 4; ++q)
      intra[tid * 4 + q] = mask[(size_t)(base + tid) * MASKW + wbase + q];
    cscore[tid] = ssc[base + tid];
    __builtin_prefetch(&mask[(size_t)(base + CHUNK + tid) * MASKW], 0, 0);
    __syncthreads();
    if (tid == 0) {
      unsigned dead[4] = { sup[wbase], sup[wbase + 1], sup[wbase + 2], sup[wbase + 3] };
      unsigned kept[4] = { 0u, 0u, 0u, 0u };
      for (int r = 0; r < CHUNK; ++r) {
        bool k = (cscore[r] > SCORE_THR) && !((dead[r >> 5] >> (r & 31)) & 1u);
        keep[base + r] = k ? 1 : 0;
        if (k) {
          kept[r >> 5] |= 1u << (r & 31);
          #pragma unroll
          for (int q = 0; q < 4; ++q) dead[q] |= intra[r * 4 + q];
        }
      }
      keptw[0] = kept[0]; keptw[1] = kept[1]; keptw[2] = kept[2]; keptw[3] = kept[3];
    }
    __syncthreads();
    // batched OR of all kept rows into the accumulator (coalesced 512B rows)
    unsigned acc = sup[tid];
    #pragma unroll 1
    for (int q = 0; q < 4; ++q) {
      unsigned kw = keptw[q];
      while (kw) {
        int r = __ffs(kw) - 1;
        kw &= kw - 1;
        acc |= mask[(size_t)(base + q * 32 + r) * MASKW + tid];
      }
    }
    sup[tid] = acc;
    __syncthreads();
  }
  for (int p = tid; p < NPRE; p += CHUNK) {
    int j = ord[p];
    comb[cls * NPRE + j] = keep[p] ? ssc[p] : -INFINITY;
  }
}

__global__ __launch_bounds__(1024)
void pp_final_topk(const float* __restrict__ comb, int* __restrict__ t500) {
  extern __shared__ unsigned long long sm[];   // 16384 * 8 = 128 KB dynamic LDS
  const int n = 16384;
  int tid = threadIdx.x;
  for (int i = tid; i < n; i += 1024) {
    if (i < NPRE * NCls)
      sm[i] = ((unsigned long long)f2s(comb[i]) << 32) | (unsigned)(~(unsigned)i);
    else
      sm[i] = 0ull;
  }
  __syncthreads();
  bitonic_desc(sm, n, tid, 1024);
  for (int i = tid; i < MAXOUT; i += 1024)
    t500[i] = (int)(~(unsigned)(sm[i] & 0xFFFFFFFFull));
}

__global__ void pp_output(const int* __restrict__ t500, const float* __restrict__ comb,
                          const float* __restrict__ box, const int* __restrict__ dcls,
                          float* __restrict__ out) {
  int i = blockIdx.x * blockDim.x + threadIdx.x;
  if (i >= MAXOUT) return;
  int flat = t500[i];
  float s = -INFINITY;
  if ((unsigned)flat < (unsigned)(NPRE * NCls)) s = comb[flat];
  else flat = 0;
  bool valid = isfinite(s);
  int label = flat / NPRE;
  int j = flat % NPRE;
  float b[7];
  #pragma unroll
  for (int k = 0; k < 7; ++k) b[k] = box[j * 7 + k];
  float d = (float)dcls[j];
  float r = b[6];
  float dir_rot = r + PI_F * 0.5f - floorf(r + 0.5f) * PI_F;
  b[6] = dir_rot - PI_F * 0.5f + PI_F * d;
  #pragma unroll
  for (int k = 0; k < 7; ++k) out[i * 7 + k] = valid ? b[k] : 0.f;
  out[MAXOUT * 7 + i] = valid ? s : 0.f;
  out[MAXOUT * 8 + i] = valid ? (float)label : -1.f;
}

// ---------------- host launcher ----------------
extern "C" void kernel_launch(void* const* d_in, const int* in_sizes, int n_in,
                              void* d_out, int out_size, void* d_ws, size_t ws_size,
                              hipStream_t stream) {
  const float* cls = (const float*)d_in[0];
  const float* bp  = (const float*)d_in[1];
  const float* dir = (const float*)d_in[2];
  const float* pri = (const float*)d_in[3];
  float* out = (float*)d_out;
  char* ws = (char*)d_ws;

  float*  maxs = (float*)(ws + O_MAXS);
  unsigned* hist = (unsigned*)(ws + O_HIST);
  int*    meta = (int*)(ws + O_META);
  unsigned* cnt = (unsigned*)(ws + O_CNT);
  unsigned long long* cand = (unsigned long long*)(ws + O_CAND);
  int*    topk = (int*)(ws + O_TOPK);
  float*  sc3  = (float*)(ws + O_SC3);
  int*    dcl  = (int*)(ws + O_DIR);
  float*  box  = (float*)(ws + O_BOX);
  float *X1 = (float*)(ws + O_X1), *Y1 = (float*)(ws + O_Y1);
  float *X2 = (float*)(ws + O_X2), *Y2 = (float*)(ws + O_Y2);
  float *AR = (float*)(ws + O_AR);
  int*    ord  = (int*)(ws + O_ORD);
  float*  ssc  = (float*)(ws + O_SSC);
  float *sx1 = (float*)(ws + O_SX1), *sy1 = (float*)(ws + O_SY1);
  float *sx2 = (float*)(ws + O_SX2), *sy2 = (float*)(ws + O_SY2);
  float *sar = (float*)(ws + O_SAR);
  int*    keep = (int*)(ws + O_KEEP);
  unsigned* mask = (unsigned*)(ws + O_MASK);
  float*  comb = (float*)(ws + O_COMB);
  int*    t500 = (int*)(ws + O_T500);

  pp_init<<<(CANDCAP + 255) / 256, 256, 0, stream>>>(hist, cnt, cand);
  pp_score_hist<<<(N + 255) / 256, 256, 0, stream>>>(cls, maxs, hist);
  pp_threshold<<<1, 1, 0, stream>>>(hist, meta);
  pp_compact<<<(N + 255) / 256, 256, 0, stream>>>(maxs, meta, cnt, cand);
  pp_sort_cand<<<1, 1024, CANDCAP * 8, stream>>>(cand, topk);
  pp_decode<<<(NPRE + 255) / 256, 256, 0, stream>>>(cls, bp, dir, pri, topk, sc3, dcl,
                                                    box, X1, Y1, X2, Y2, AR);
  for (int c = 0; c < NCls; ++c) {
    pp_sort_class<<<1, 1024, 0, stream>>>(sc3, X1, Y1, X2, Y2, AR,
                                          ord, ssc, sx1, sy1, sx2, sy2, sar, c);
    pp_nms_mask<<<dim3(NPRE / 32, NPRE / 16), 32, 0, stream>>>(sx1, sy1, sx2, sy2, sar, mask);
    pp_nms_scan<<<1, CHUNK, 0, stream>>>(ssc, ord, mask, keep, comb, c);
  }
  pp_final_topk<<<1, 1024, 16384 * 8, stream>>>(comb, t500);
  pp_output<<<(MAXOUT + 255) / 256, 256, 0, stream>>>(t500, comb, box, dcl, out);
}